// RelationalAttentionContextEncoder2_27127013441943
// MI455X (gfx1250) — compile-verified
//
#include <hip/hip_runtime.h>
#include <hip/hip_bf16.h>

// CDNA5 / gfx1250, wave32.
// rel GEMM: feats[B*256, 12] @ W_rel'[12,128] via V_WMMA_F32_16X16X4_F32.
// K=0..8 real features, K=9 = bias indicator (1.0) with W[9]=b_rel,
// K=10,11 = zero pad. Diagonal row (j==i) zeroed in the A fragment so it
// contributes relu(0)=0 to the pool -> pooling is a plain row sum.
// relu on WMMA outputs via inline-asm v_max_num_f32: clang inserts a
// canonicalize (max x,x) before any IR-level maxnum/med3 on non-arithmetic
// values; the asm form is opaque to instcombine and emits exactly one op.

typedef float v2f __attribute__((ext_vector_type(2)));
typedef float v8f __attribute__((ext_vector_type(8)));

#define HDIM   128
#define NBATCH 8192

__device__ __forceinline__ float relu_raw(float x) {
    float r;
    // VOP2: src0 may be an inline constant, src1 must be VGPR.
    asm("v_max_num_f32 %0, 0, %1" : "=v"(r) : "v"(x));
    return r;
}

__global__ __launch_bounds__(512) void relctx_wmma_kernel(
    const float* __restrict__ ctx,     // [B, 64]
    const float* __restrict__ W_rel,   // [9, 128] row-major
    const float* __restrict__ b_rel,   // [128]
    const float* __restrict__ W_prop,  // [2, 128]
    const float* __restrict__ b_prop,  // [128]
    float* __restrict__ out)           // [B, 16, 256]
{
    const int lane = threadIdx.x & 31;
    const int i    = threadIdx.x >> 5;   // wave id == entity index i (0..15)
    const bool hi  = lane >= 16;         // upper lane-half
    const int col  = lane & 15;          // column within 16-wide tile; also j index

    // ---- Preload W_rel as WMMA B-fragments ------------------------------
    // B layout (16x16x4, 32-bit): lane = N column; lo lanes hold K=k0,k0+1 in
    // vgpr0,vgpr1; hi lanes hold K=k0+2,k0+3.
    float b0s0[8], b1s0[8], b0s1[8], b1s1[8], b0s2[8];
    float brel[8], wp0[8], wp1[8], bp[8];
#pragma unroll
    for (int n = 0; n < 8; ++n) {
        const int c = n * 16 + col;
        b0s0[n] = W_rel[(hi ? 2 : 0) * HDIM + c];
        b1s0[n] = W_rel[(hi ? 3 : 1) * HDIM + c];
        b0s1[n] = W_rel[(hi ? 6 : 4) * HDIM + c];
        b1s1[n] = W_rel[(hi ? 7 : 5) * HDIM + c];
        b0s2[n] = hi ? 0.0f : W_rel[8 * HDIM + c];  // K=8 | K=10 (pad)
        brel[n] = b_rel[c];                          // K=9 bias row (lo half)
        wp0[n]  = W_prop[c];
        wp1[n]  = W_prop[HDIM + c];
        bp[n]   = b_prop[c];
    }

    // per-lane output offset: lanes 0-15 write prop[0:128], 16-31 write rel[128:256]
    const int lane_off = (hi ? HDIM : 0) + col;

    for (int b = blockIdx.x; b < NBATCH; b += gridDim.x) {
        const float4* erow = (const float4*)(ctx + (size_t)b * 64);
        const float4 ej = erow[col];   // entity j = col (same in both halves)
        const float4 ei = erow[i];     // uniform per wave
        const float d0 = ei.x - ej.x, d1 = ei.y - ej.y;
        const float d2 = ei.z - ej.z, d3 = ei.w - ej.w;
        const float dist = __builtin_amdgcn_sqrtf(d0 * d0 + d1 * d1);

        // A fragments: lane = row j. Diagonal row (j==i) forced to all-zero,
        // including the bias indicator, so its relu output is exactly 0.
        const bool zd = (col == i);
        v2f a0, a1, a2;
        a0.x = zd ? 0.0f : (hi ? ej.z : ei.z);          // K 0 | 2
        a0.y = zd ? 0.0f : (hi ? ej.w : ei.w);          // K 1 | 3
        a1.x = zd ? 0.0f : (hi ? d2 : d0);              // K 4 | 6
        a1.y = zd ? 0.0f : (hi ? d3 : d1);              // K 5 | 7
        a2.x = (zd || hi) ? 0.0f : dist;                // K 8 | 10(pad)
        a2.y = (zd || hi) ? 0.0f : 1.0f;                // K 9 (bias ind.) | 11(pad)

        float* olane = out + (size_t)b * 4096 + i * 256 + lane_off;

#pragma unroll
        for (int n = 0; n < 8; ++n) {
            v2f bb0; bb0.x = b0s0[n]; bb0.y = b1s0[n];
            v2f bb1; bb1.x = b0s1[n]; bb1.y = b1s1[n];
            v2f bb2; bb2.x = b0s2[n]; bb2.y = hi ? 0.0f : brel[n];  // W[8] ; b_rel

            v8f acc = {};
            acc = __builtin_amdgcn_wmma_f32_16x16x4_f32(false, a0, false, bb0,
                                                        (short)0, acc, false, false);
            acc = __builtin_amdgcn_wmma_f32_16x16x4_f32(false, a1, false, bb1,
                                                        (short)0, acc, false, false);
            acc = __builtin_amdgcn_wmma_f32_16x16x4_f32(false, a2, false, bb2,
                                                        (short)0, acc, false, false);

            // relu (1 op/elem via asm) + pool over all 16 rows; pairwise
            // float2 adds so the tree can lower to v_pk_add_f32.
            v2f p0, p1, p2, p3;
            p0.x = relu_raw(acc[0]); p0.y = relu_raw(acc[1]);
            p1.x = relu_raw(acc[2]); p1.y = relu_raw(acc[3]);
            p2.x = relu_raw(acc[4]); p2.y = relu_raw(acc[5]);
            p3.x = relu_raw(acc[6]); p3.y = relu_raw(acc[7]);
            v2f t = (p0 + p1) + (p2 + p3);
            float part = t.x + t.y;
            part += __shfl_xor(part, 16, 32);            // merge row-halves
            const float pooled = part * (1.0f / 15.0f);

            // prop head: FMA result is canonical -> fmaxf emits a single max
            const float pr = fmaxf(fmaf(ei.z, wp0[n], fmaf(ei.w, wp1[n], bp[n])), 0.0f);

            // full-wave store: lo half -> prop column, hi half -> pooled column
            olane[n * 16] = hi ? pooled : pr;
        }
    }
}

extern "C" void kernel_launch(void* const* d_in, const int* in_sizes, int n_in,
                              void* d_out, int out_size, void* d_ws, size_t ws_size,
                              hipStream_t stream) {
    const float* ctx    = (const float*)d_in[0];
    const float* W_rel  = (const float*)d_in[1];
    const float* b_rel  = (const float*)d_in[2];
    const float* W_prop = (const float*)d_in[3];
    const float* b_prop = (const float*)d_in[4];
    float* out = (float*)d_out;

    // 2048 blocks x 512 threads (16 waves/block, wave == entity i),
    // batch grid-stride (4 items per block) amortizes the W preloads.
    relctx_wmma_kernel<<<dim3(2048), dim3(512), 0, stream>>>(
        ctx, W_rel, b_rel, W_prop, b_prop, out);
}